// EnergyBasedVAD_17927193493860
// MI455X (gfx1250) — compile-verified
//
#include <hip/hip_runtime.h>
#include <hip/hip_bf16.h>
#include <stdint.h>

// ---------------------------------------------------------------------------
// EnergyBasedVAD for MI455X (gfx1250)
//
// Roofline: dominant cost is reading the waveform: 32*960000*4 B = 122.9 MB.
// At 23.3 TB/s HBM that's ~5.3 us; FLOPs (~123 MF) are irrelevant. So the
// energy kernel is built around data movement: TDM async global->LDS staging
// (tensor_load_to_lds + s_wait_tensorcnt), and the 256-wide per-hop
// sum-of-squares is done on the matrix pipe with V_WMMA_F32_16X16X4_F32
// (B = ones => every column of D is the row sum), keeping VALU free.
// Overlapping frames are handled by computing per-hop (256-sample) partial
// sums once: energy[f] = (hop[f] + hop[f+1]) / 512.
// ---------------------------------------------------------------------------

typedef uint32_t u32;
typedef uint64_t u64;
typedef u32   u32x4 __attribute__((ext_vector_type(4)));
typedef int   i32x4 __attribute__((ext_vector_type(4)));
typedef int   i32x8 __attribute__((ext_vector_type(8)));
typedef float v2f   __attribute__((ext_vector_type(2)));
typedef float v8f   __attribute__((ext_vector_type(8)));

#define HOP 256
#define FRAME 512
#define TILE_HOPS 32                 // hops per block
#define TILE_ELEMS (TILE_HOPS * HOP) // 8192 floats per TDM tile
#define LDS_STRIDE 258               // 256 data dwords + 2 pad dwords per hop
#define SILENCE_FRAMES 18            // int(0.3*16000/256)
#define MIN_SPEECH_FRAMES 6          // int(0.1*16000/256)

// ---------------------------------------------------------------------------
// Kernel 1: hopsum[b][h] = sum_{i<256} wav[b][h*256+i]^2
// 64 threads (2 waves). TDM stages a 32-hop tile into LDS (padded stride),
// each wave reduces 16 hop rows with chained WMMA f32 16x16x4 (B = ones).
// ---------------------------------------------------------------------------
__global__ void vad_hopsum_kernel(const float* __restrict__ wav,
                                  float* __restrict__ hopsum,
                                  int N, int HOPS) {
    extern __shared__ float lds[];
    const int b    = blockIdx.y;
    const int hop0 = blockIdx.x * TILE_HOPS;

    // --- TDM: async copy tile global -> LDS, wave 0 only (TDM ignores EXEC,
    // one issue per wave). Padding: every 256 dwords insert 2 dwords so hop h
    // starts at lds[h*258] (8B aligned, bank-conflict-friendly). OOB reads
    // past tensor_dim0 return zero -> ragged tail auto-zero-filled.
    if ((threadIdx.x >> 5) == 0) {
        u64 gaddr = (u64)(const void*)(wav + (u64)b * N + (u64)hop0 * HOP);
        u32 remaining = (u32)(N - hop0 * HOP);  // > 0 by grid construction

        u32x4 g0;
        g0[0] = 1u;                                  // count=1, user mode
        g0[1] = 0u;                                  // lds_addr = 0 (dynamic LDS base)
        g0[2] = (u32)(gaddr & 0xFFFFFFFFull);        // global_addr[31:0]
        g0[3] = (u32)((gaddr >> 32) & 0x1FFFFFFull)  // global_addr[56:32]
              | (2u << 30);                          // type = 2 (image/D#)

        i32x8 g1;
        g1[0] = (int)((2u << 16)      // data_size = 4B
                    | (1u << 20)      // pad_enable
                    | (7u << 22)      // pad_interval = 256 dwords
                    | (1u << 25));    // pad_amount  = 2 dwords
        g1[1] = (int)((remaining & 0xFFFFu) << 16);          // tensor_dim0[15:0]
        g1[2] = (int)((remaining >> 16) | (1u << 16));       // tensor_dim0[31:16] | tensor_dim1=1
        g1[3] = (int)((u32)TILE_ELEMS << 16);                // tile_dim0 = 8192
        g1[4] = (int)1;                                      // tile_dim1 = 1, tile_dim2 = 0
        g1[5] = (int)remaining;                              // tensor_dim0_stride (unused, Y=1)
        g1[6] = 0;
        g1[7] = 0;

        i32x4 gz4 = {0, 0, 0, 0};
        i32x8 gz8 = {0, 0, 0, 0, 0, 0, 0, 0};
        __builtin_amdgcn_tensor_load_to_lds(g0, g1, gz4, gz4, gz8, 0);
        __builtin_amdgcn_s_wait_tensorcnt(0);
    }
    __syncthreads();

    // --- WMMA row-sum: wave w owns hop rows [w*16, w*16+16).
    // A (16x4 f32, 2 VGPRs): lanes 0-15 carry K0/K1, lanes 16-31 carry K2/K3
    // of the current 4-wide K window; B = ones so D[m][n] = rowsum(m) for all n.
    const int lane = threadIdx.x & 31;
    const int w    = threadIdx.x >> 5;        // wave 0..1
    const int row  = lane & 15;
    const int khi  = lane >> 4;               // 0: K0/K1, 1: K2/K3
    const float* base = &lds[(w * 16 + row) * LDS_STRIDE + 2 * khi];

    v8f acc = {0.f, 0.f, 0.f, 0.f, 0.f, 0.f, 0.f, 0.f};
    v2f ones = {1.0f, 1.0f};
    #pragma unroll 4
    for (int i = 0; i < 64; ++i) {            // 64 * K4 = 256 samples
        v2f x = *(const v2f*)(base + 4 * i);  // 8B aligned: stride 258 is even
        v2f a;
        a.x = x.x * x.x;
        a.y = x.y * x.y;
        acc = __builtin_amdgcn_wmma_f32_16x16x4_f32(
                  false, a, false, ones, (short)0, acc, false, false);
    }

    // D layout: VGPR r, lanes 0-15 -> M=r (col=lane), lanes 16-31 -> M=r+8.
    // All columns equal; lane 0 writes rows 0..7, lane 16 writes rows 8..15.
    if ((lane & 15) == 0) {
        const int mbase = w * 16 + khi * 8;
        #pragma unroll
        for (int r = 0; r < 8; ++r) {
            const int h = hop0 + mbase + r;
            if (h < HOPS) hopsum[(u64)b * HOPS + h] = acc[r];
        }
    }
}

// ---------------------------------------------------------------------------
// Kernel 2: per-batch energies + 20th-percentile threshold (bitonic sort in LDS)
// ---------------------------------------------------------------------------
__global__ void vad_threshold_kernel(const float* __restrict__ hopsum,
                                     float* __restrict__ energy,
                                     float* __restrict__ thr,
                                     int HOPS, int FRAMES) {
    __shared__ float s[4096];
    const int b = blockIdx.x;
    const float* hs = hopsum + (u64)b * HOPS;
    float* eb = energy + (u64)b * FRAMES;

    for (int f = threadIdx.x; f < 4096; f += blockDim.x) {
        float v = 3.3e38f;  // pad sorts to the top, never selected (quantile < FRAMES)
        if (f < FRAMES) {
            v = (hs[f] + hs[f + 1]) * (1.0f / (float)FRAME);
            eb[f] = v;
        }
        s[f] = v;
    }
    __syncthreads();

    // Bitonic sort, 4096 elements. Each (k,j) pass: every element belongs to
    // exactly one compare pair, handled by the lower index -> race-free.
    for (int k = 2; k <= 4096; k <<= 1) {
        for (int j = k >> 1; j > 0; j >>= 1) {
            for (int q = threadIdx.x; q < 4096; q += blockDim.x) {
                const int p = q ^ j;
                if (p > q) {
                    const float a = s[q], c = s[p];
                    const bool up = ((q & k) == 0);
                    if (up ? (a > c) : (a < c)) { s[q] = c; s[p] = a; }
                }
            }
            __syncthreads();
        }
    }

    if (threadIdx.x == 0) {
        // nzero = #(e <= 0); sorted -> first index with s[i] > 0
        int lo = 0, hi = FRAMES;
        while (lo < hi) {
            const int mid = (lo + hi) >> 1;
            if (s[mid] > 0.0f) hi = mid; else lo = mid + 1;
        }
        const int nzero = lo;
        const int nz = FRAMES - nzero;
        float t;
        if (nz > 0) {
            const float pos = 0.2f * (float)(nz - 1);
            const int plo = (int)floorf(pos);
            const int phi = (int)ceilf(pos);
            const float frac = pos - (float)plo;
            int ilo = nzero + plo; ilo = ilo < 0 ? 0 : (ilo > FRAMES - 1 ? FRAMES - 1 : ilo);
            int ihi = nzero + phi; ihi = ihi < 0 ? 0 : (ihi > FRAMES - 1 ? FRAMES - 1 : ihi);
            t = s[ilo] * (1.0f - frac) + s[ihi] * frac;
        } else {
            t = 0.01f;  // ENERGY_THRESHOLD
        }
        thr[b] = t;
    }
}

// ---------------------------------------------------------------------------
// Kernel 3: per-batch sequential VAD state machine + segment painting.
// Block clears its output row (d_out is poisoned by the harness), then
// thread 0 runs the 3749-step scan (inherently sequential, trivial cost).
// ---------------------------------------------------------------------------
__global__ void vad_scan_kernel(const float* __restrict__ energy,
                                const float* __restrict__ thr,
                                float* __restrict__ out,
                                int FRAMES) {
    const int b = blockIdx.x;
    const float* eb = energy + (u64)b * FRAMES;
    float* ob = out + (u64)b * FRAMES;

    for (int i = threadIdx.x; i < FRAMES; i += blockDim.x) ob[i] = 0.0f;
    __syncthreads();

    if (threadIdx.x == 0) {
        const float t = thr[b];
        bool in_sp = false;
        int start = 0, sil = 0;
        for (int f = 0; f < FRAMES; ++f) {
            const bool mt = eb[f] > t;
            if (mt && !in_sp) start = f;
            sil = mt ? 0 : (in_sp ? sil + 1 : sil);
            const bool closed = !mt && in_sp && (sil >= SILENCE_FRAMES);
            if (closed) {
                const int end = f - sil;
                if (end - start >= MIN_SPEECH_FRAMES)
                    for (int i = start; i < end; ++i) ob[i] = 1.0f;
                in_sp = false;
            }
            if (mt) in_sp = true;
        }
        if (in_sp && (FRAMES - start >= MIN_SPEECH_FRAMES))
            for (int i = start; i < FRAMES; ++i) ob[i] = 1.0f;
    }
}

// ---------------------------------------------------------------------------
extern "C" void kernel_launch(void* const* d_in, const int* in_sizes, int n_in,
                              void* d_out, int out_size, void* d_ws, size_t ws_size,
                              hipStream_t stream) {
    const float* wav = (const float*)d_in[0];
    float* out = (float*)d_out;
    float* ws  = (float*)d_ws;

    const int B = 32;                       // reference batch
    const int N = in_sizes[0] / B;          // 960000
    const int HOPS = N / HOP;               // 3750
    const int FRAMES = (N - FRAME) / HOP + 1; // 3749

    float* hopsum = ws;                     // B*HOPS floats
    float* energy = ws + (size_t)B * HOPS;  // B*FRAMES floats
    float* thr    = energy + (size_t)B * FRAMES; // B floats

    // Kernel 1: 2 waves/block, 32 hops/block, 33 KB LDS (32 * 258 * 4).
    dim3 g1((HOPS + TILE_HOPS - 1) / TILE_HOPS, B);
    vad_hopsum_kernel<<<g1, dim3(64), TILE_HOPS * LDS_STRIDE * sizeof(float), stream>>>(
        wav, hopsum, N, HOPS);

    vad_threshold_kernel<<<dim3(B), dim3(512), 0, stream>>>(hopsum, energy, thr, HOPS, FRAMES);

    vad_scan_kernel<<<dim3(B), dim3(256), 0, stream>>>(energy, thr, out, FRAMES);
}